// GraphConvolution_GCN_50105088475682
// MI455X (gfx1250) — compile-verified
//
#include <hip/hip_runtime.h>
#include <hip/hip_bf16.h>

typedef float v2f __attribute__((ext_vector_type(2)));
typedef float v8f __attribute__((ext_vector_type(8)));

#define IN_F     128
#define OUT_F    256
#define N_NODES  50000
#define N_EDGES  800000
#define LDS_STRIDE 132      // 128 + 4 pad: row reads hit distinct bank groups
#define MT       4          // M-tiles per wave (64 rows per block)
#define ROWS_PB  (16 * MT)  // 64

// ---------------------------------------------------------------------------
// Kernel 1: zero the aggregation workspace (N_NODES * IN_F floats)
// ---------------------------------------------------------------------------
__global__ void gcn_zero_kernel(float4* __restrict__ agg4, int n4) {
    int i = blockIdx.x * blockDim.x + threadIdx.x;
    if (i < n4) agg4[i] = make_float4(0.f, 0.f, 0.f, 0.f);
}

// ---------------------------------------------------------------------------
// Kernel 2: SpMM scatter.  One wave per edge; edge metadata is wave-uniform
// (forced through readfirstlane -> scalar loads).  Each lane handles 4
// features: float4 gather of x[col] row is one fully-coalesced 512B row per
// wave; scatter via global_atomic_add_f32 into L2-resident agg.
// ---------------------------------------------------------------------------
__global__ __launch_bounds__(256)
void gcn_spmm_kernel(const float* __restrict__ x,
                     const int*   __restrict__ edge_row,
                     const int*   __restrict__ edge_col,
                     const float* __restrict__ edge_val,
                     float*       __restrict__ agg,
                     int n_edges) {
    const int lane = threadIdx.x & 31;
    int e = __builtin_amdgcn_readfirstlane(blockIdx.x * 8 + (threadIdx.x >> 5));
    if (e >= n_edges) return;

    const int   r = edge_row[e];   // scalar (uniform address)
    const int   c = edge_col[e];
    const float v = edge_val[e];

    const float4 xv = ((const float4*)(x + (size_t)c * IN_F))[lane];
    float* dst = agg + (size_t)r * IN_F + lane * 4;
    atomicAdd(dst + 0, v * xv.x);
    atomicAdd(dst + 1, v * xv.y);
    atomicAdd(dst + 2, v * xv.z);
    atomicAdd(dst + 3, v * xv.w);
}

// ---------------------------------------------------------------------------
// Kernel 3: fp32 GEMM  out = agg[50000x128] @ W[128x256] + b  via
// V_WMMA_F32_16X16X4_F32.  Block = 8 waves, block tile = 64 rows x 128 cols.
// Each wave: one 16-col slice, MT=4 stacked 16x16 tiles -> each B register
// pair feeds 4 WMMAs; A tile staged once in LDS and reused by all 8 waves.
// Tail-block row guards are hoisted to a single block-level branch so the
// common path has unpredicated loads/stores with immediate offsets.
//
// ISA VGPR layouts (CDNA5 ISA):
//   A 16x4 f32 : lane L -> M = L%16 ; regs = K = koff, koff+1, koff=(L<16?0:2)
//   B 4x16 f32 : symmetric (K pairs in regs, N = L%16 across lanes)
//   C/D 16x16  : VGPR i -> M = i + (L>=16 ? 8 : 0), N = L%16
// ---------------------------------------------------------------------------
__global__ __launch_bounds__(256)
void gcn_gemm_kernel(const float* __restrict__ agg,
                     const float* __restrict__ W,     // [IN_F][OUT_F] row-major
                     const float* __restrict__ bias,  // [OUT_F]
                     float*       __restrict__ out) { // [N_NODES][OUT_F]
    __shared__ float Atile[ROWS_PB * LDS_STRIDE];     // 64*132*4 = 33 KB

    const int  row0    = blockIdx.x * ROWS_PB;
    const int  colBase = blockIdx.y * 128;
    const int  tid     = threadIdx.x;
    const int  wave    = tid >> 5;
    const int  lane    = tid & 31;
    const bool full    = (row0 + ROWS_PB) <= N_NODES;   // block-uniform

    // Cooperative A stage: 64 rows x 128 floats (float4 flat copy, padded
    // LDS stride).  Guarded zero-fill only in the single tail block.
    if (full) {
        const float4* src = (const float4*)(agg + (size_t)row0 * IN_F);
#pragma unroll
        for (int it = 0; it < 8; ++it) {
            const int f4 = it * 256 + tid;          // 0..2047 float4s
            const int r  = f4 >> 5;
            const int c4 = f4 & 31;
            *(float4*)(Atile + r * LDS_STRIDE + c4 * 4) = src[f4];
        }
    } else {
#pragma unroll
        for (int it = 0; it < 8; ++it) {
            const int f4 = it * 256 + tid;
            const int r  = f4 >> 5;
            const int c4 = f4 & 31;
            float4 val = make_float4(0.f, 0.f, 0.f, 0.f);
            if (row0 + r < N_NODES)
                val = ((const float4*)(agg + (size_t)(row0 + r) * IN_F))[c4];
            *(float4*)(Atile + r * LDS_STRIDE + c4 * 4) = val;
        }
    }
    __syncthreads();

    const bool hi    = lane >= 16;
    const int  nsel  = lane & 15;
    const int  myCol = colBase + wave * 16 + nsel;
    const int  koff  = hi ? 2 : 0;

    const float* Bcol  = W + (size_t)koff * OUT_F + myCol;
    const float* Abase = Atile + nsel * LDS_STRIDE + koff;

    v8f acc[MT];
#pragma unroll
    for (int t = 0; t < MT; ++t)
        acc[t] = (v8f){0.f, 0.f, 0.f, 0.f, 0.f, 0.f, 0.f, 0.f};

#pragma unroll 4
    for (int kk = 0; kk < IN_F / 4; ++kk) {
        const int k = kk * 4;
        v2f b;
        b.x = Bcol[(size_t)k * OUT_F];           // W[k+koff  ][myCol]
        b.y = Bcol[(size_t)(k + 1) * OUT_F];     // W[k+koff+1][myCol]
#pragma unroll
        for (int t = 0; t < MT; ++t) {
            const float2 af =
                *(const float2*)(Abase + t * 16 * LDS_STRIDE + k);
            v2f a; a.x = af.x; a.y = af.y;
            acc[t] = __builtin_amdgcn_wmma_f32_16x16x4_f32(
                false, a, false, b, (short)0, acc[t], false, false);
        }
    }

    // Bias + store.  VGPR i of tile t -> row (row0 + t*16 + i + hi*8).
    const float bv = bias[myCol];
    if (full) {
        // Fast path: base pointer + immediate offsets, no predication.
        float* obase = out + (size_t)(row0 + (hi ? 8 : 0)) * OUT_F + myCol;
#pragma unroll
        for (int t = 0; t < MT; ++t) {
            float* o = obase + (size_t)t * 16 * OUT_F;
#pragma unroll
            for (int i = 0; i < 8; ++i)
                o[(size_t)i * OUT_F] = acc[t][i] + bv;
        }
    } else {
#pragma unroll
        for (int t = 0; t < MT; ++t) {
            const int mbase = row0 + t * 16 + (hi ? 8 : 0);
#pragma unroll
            for (int i = 0; i < 8; ++i) {
                const int row = mbase + i;
                if (row < N_NODES)
                    out[(size_t)row * OUT_F + myCol] = acc[t][i] + bv;
            }
        }
    }
}

// ---------------------------------------------------------------------------
extern "C" void kernel_launch(void* const* d_in, const int* in_sizes, int n_in,
                              void* d_out, int out_size, void* d_ws, size_t ws_size,
                              hipStream_t stream) {
    const float* x        = (const float*)d_in[0];
    const int*   edge_row = (const int*)  d_in[1];
    const int*   edge_col = (const int*)  d_in[2];
    const float* edge_val = (const float*)d_in[3];
    const float* weight   = (const float*)d_in[4];
    const float* bias     = (const float*)d_in[5];
    float*       out      = (float*)d_out;
    float*       agg      = (float*)d_ws;    // N_NODES * IN_F floats = 25.6 MB

    // 1) zero workspace
    {
        int n4 = N_NODES * IN_F / 4;
        gcn_zero_kernel<<<(n4 + 255) / 256, 256, 0, stream>>>((float4*)agg, n4);
    }
    // 2) SpMM scatter: one wave (32 lanes) per edge, 8 edges per block
    {
        int blocks = (N_EDGES + 7) / 8;
        gcn_spmm_kernel<<<blocks, 256, 0, stream>>>(x, edge_row, edge_col,
                                                    edge_val, agg, N_EDGES);
    }
    // 3) WMMA GEMM + bias: 64-row x 128-col block tiles
    {
        dim3 grid((N_NODES + ROWS_PB - 1) / ROWS_PB, OUT_F / 128);  // 782 x 2
        gcn_gemm_kernel<<<grid, 256, 0, stream>>>(agg, weight, bias, out);
    }
}